// Net_88510686036594
// MI455X (gfx1250) — compile-verified
//
#include <hip/hip_runtime.h>
#include <hip/hip_bf16.h>

// ---------------------------------------------------------------------------
// RGCN layer for MI455X (gfx1250):
//   out = X @ root + bias + sum_r (per-(dst,rel) mean of X[src]) @ W_r
// Strategy: H = X @ [W_0..W_7 | root]  (50000x128x144 f32 WMMA GEMM),
// then per-edge scatter of 16-float rows with 1/cnt scaling.
// ---------------------------------------------------------------------------

#define N_REL   8
#define D_IN    128
#define D_OUT   16
#define D_CAT   144            // 8*16 relation cols + 16 root cols

typedef __attribute__((ext_vector_type(2))) float v2f;
typedef __attribute__((ext_vector_type(8))) float v8f;

// ---- build Bfull[128][144]: cols [r*16+o] = W[r][d][o], cols [128+o] = root[d][o]
__global__ void k_build_B(const float* __restrict__ W,
                          const float* __restrict__ root,
                          float* __restrict__ Bfull) {
    int tid = blockIdx.x * blockDim.x + threadIdx.x;
    if (tid >= D_IN * D_CAT) return;
    int d = tid / D_CAT;
    int c = tid % D_CAT;
    float v;
    if (c < N_REL * D_OUT) {
        int r = c >> 4, o = c & 15;
        v = W[((size_t)r * D_IN + d) * D_OUT + o];
    } else {
        v = root[(size_t)d * D_OUT + (c - N_REL * D_OUT)];
    }
    Bfull[tid] = v;
}

// ---- zero the (dst, rel) edge-count buffer
__global__ void k_zero_cnt(int* __restrict__ cnt, int n) {
    int tid = blockIdx.x * blockDim.x + threadIdx.x;
    if (tid < n) cnt[tid] = 0;
}

// ---- count edges per (dst, rel)
__global__ void k_count(const int* __restrict__ edge_index,
                        const int* __restrict__ edge_type,
                        int* __restrict__ cnt, int E) {
    int e = blockIdx.x * blockDim.x + threadIdx.x;
    if (e >= E) return;
    int d = edge_index[E + e];
    int r = edge_type[e];
    atomicAdd(&cnt[d * N_REL + r], 1);
}

// ---- H[N,144] = X[N,128] @ Bfull[128,144] via V_WMMA_F32_16X16X4_F32
// One wave (32 lanes) per 16x16 output tile; 32 K-steps of 4.
// A 16x4 frag : lane L -> M = L%16, K = 2*(L/16) + v   (v = VGPR index)
// B  4x16 frag: lane L -> N = L%16, K = 2*(L/16) + v
// C/D 16x16   : lane L -> N = L%16, M = v + 8*(L/16)
__global__ __launch_bounds__(32) void k_gemm_wmma(const float* __restrict__ X,
                                                  const float* __restrict__ B,
                                                  float* __restrict__ H) {
    const int row0 = blockIdx.x * 16;           // node-tile
    const int col0 = blockIdx.y * 16;           // column-tile of Bfull/H
    const int lane = threadIdx.x;               // 0..31, full wave
    const int m    = lane & 15;
    const int half = lane >> 4;

    v8f acc = {};
    const float* xrow = X + (size_t)(row0 + m) * D_IN;     // row for A's M
    const float* bcol = B + col0 + m;                      // column for B's N

    #pragma unroll 8
    for (int k0 = 0; k0 < D_IN; k0 += 4) {
        const int ka = k0 + 2 * half;
        v2f a, b;
        a.x = xrow[ka];
        a.y = xrow[ka + 1];
        b.x = bcol[(size_t)ka * D_CAT];
        b.y = bcol[(size_t)(ka + 1) * D_CAT];
        acc = __builtin_amdgcn_wmma_f32_16x16x4_f32(
            /*neg_a=*/false, a, /*neg_b=*/false, b,
            /*c_mod=*/(short)0, acc, /*reuse_a=*/false, /*reuse_b=*/false);
    }

    #pragma unroll
    for (int v = 0; v < 8; ++v) {
        const int r = row0 + v + 8 * half;
        H[(size_t)r * D_CAT + col0 + m] = acc[v];
    }
}

// ---- out[n][o] = H[n][128+o] + bias[o]   (root term + bias)
__global__ void k_init_out(const float* __restrict__ H,
                           const float* __restrict__ bias,
                           float* __restrict__ out, int N) {
    int tid = blockIdx.x * blockDim.x + threadIdx.x;
    if (tid >= N * D_OUT) return;
    int n = tid >> 4, o = tid & 15;
    out[tid] = H[(size_t)n * D_CAT + N_REL * D_OUT + o] + bias[o];
}

// ---- per-edge scatter: out[dst][o] += H[src][r*16+o] / cnt[dst][r]
// 16 consecutive lanes handle one edge -> coalesced 64B atomic bursts.
__global__ void k_aggregate(const int* __restrict__ edge_index,
                            const int* __restrict__ edge_type,
                            const int* __restrict__ cnt,
                            const float* __restrict__ H,
                            float* __restrict__ out, int E) {
    int tid = blockIdx.x * blockDim.x + threadIdx.x;
    int e = tid >> 4;
    int o = tid & 15;
    if (e >= E) return;
    int s = edge_index[e];
    int d = edge_index[E + e];
    int r = edge_type[e];
    float scale = 1.0f / (float)cnt[d * N_REL + r];   // cnt >= 1 for any live edge
    float v = H[(size_t)s * D_CAT + r * D_OUT + o] * scale;
    unsafeAtomicAdd(&out[d * D_OUT + o], v);
}

extern "C" void kernel_launch(void* const* d_in, const int* in_sizes, int n_in,
                              void* d_out, int out_size, void* d_ws, size_t ws_size,
                              hipStream_t stream) {
    // inputs per setup_inputs() order
    const float* x          = (const float*)d_in[0];   // [N, 128]
    const float* W          = (const float*)d_in[1];   // [8, 128, 16]
    const float* root       = (const float*)d_in[2];   // [128, 16]
    const float* bias       = (const float*)d_in[3];   // [16]
    const int*   edge_index = (const int*)d_in[4];     // [2, E] flat
    const int*   edge_type  = (const int*)d_in[5];     // [E]
    // d_in[6] = edge_ptr (unused; edge_type gives the relation directly)

    const int N = in_sizes[0] / D_IN;                  // 50000
    const int E = in_sizes[4] / 2;                     // 600000
    float* out = (float*)d_out;                        // [N, 16]

    // workspace carve-up
    char*  ws    = (char*)d_ws;
    float* H     = (float*)ws;                                    // N*144 f32
    size_t offH  = (size_t)N * D_CAT * sizeof(float);
    float* Bfull = (float*)(ws + offH);                           // 128*144 f32
    size_t offB  = offH + (size_t)D_IN * D_CAT * sizeof(float);
    int*   cnt   = (int*)(ws + offB);                             // N*8 i32

    const int TB = 256;

    // 1) pack [W | root] into Bfull[128][144]
    k_build_B<<<(D_IN * D_CAT + TB - 1) / TB, TB, 0, stream>>>(W, root, Bfull);

    // 2) per-(dst,rel) edge counts
    k_zero_cnt<<<(N * N_REL + TB - 1) / TB, TB, 0, stream>>>(cnt, N * N_REL);
    k_count<<<(E + TB - 1) / TB, TB, 0, stream>>>(edge_index, edge_type, cnt, E);

    // 3) H = X @ Bfull  (3125 x 9 tiles of 16x16, one wave each)
    dim3 ggrid(N / 16, D_CAT / 16);
    k_gemm_wmma<<<ggrid, 32, 0, stream>>>(x, Bfull, H);

    // 4) out = root-term + bias
    k_init_out<<<(N * D_OUT + TB - 1) / TB, TB, 0, stream>>>(H, bias, out, N);

    // 5) scatter relation means into out
    long long athreads = (long long)E * D_OUT;
    k_aggregate<<<(int)((athreads + TB - 1) / TB), TB, 0, stream>>>(
        edge_index, edge_type, cnt, H, out, E);
}